// GTG_Module_85744727098082
// MI455X (gfx1250) — compile-verified
//
#include <hip/hip_runtime.h>
#include <hip/hip_bf16.h>
#include <math.h>

typedef __attribute__((ext_vector_type(2))) float v2f;
typedef __attribute__((ext_vector_type(8))) float v8f;

#define N 4096
#define D 1024
#define C 100
#define CPAD 128          // padded col count for X^T (zero rows 100..127)
#define GTG_ITERS 30
#define NLAB 2048
#define EPS 1e-8f

// ---------------- ws layout (bytes) ----------------
#define OFF_EN    ((size_t)0)                       // 4096*1024*4 = 16 MB
#define OFF_A     ((size_t)16777216)                // 4096*4096*4 = 64 MB
#define OFF_SIG   (OFF_A + (size_t)67108864)        // 4096*4
#define OFF_PART  (OFF_SIG + (size_t)16384)         // 16384*4
#define OFF_MEAN  (OFF_PART + (size_t)65536)        // 4 (64B slot)
#define OFF_X     (OFF_MEAN + (size_t)64)           // 4096*100*4
#define OFF_Y     (OFF_X + (size_t)1638400)         // 4096*100*4
#define OFF_XT    (OFF_Y + (size_t)1638400)         // 128*4096*4 = 2 MB

// ============ 1) row-normalize embedding ============
__global__ void k_normalize(const float* __restrict__ E, float* __restrict__ En) {
    __shared__ float red[256];
    int row = blockIdx.x, tid = threadIdx.x;
    float v[4]; float ss = 0.f;
    #pragma unroll
    for (int e = 0; e < 4; ++e) {
        v[e] = E[(size_t)row * D + e * 256 + tid];
        ss += v[e] * v[e];
    }
    red[tid] = ss; __syncthreads();
    for (int s = 128; s >= 1; s >>= 1) {
        if (tid < s) red[tid] += red[tid + s];
        __syncthreads();
    }
    float inv = 1.0f / fmaxf(sqrtf(red[0]), 1e-12f);
    #pragma unroll
    for (int e = 0; e < 4; ++e)
        En[(size_t)row * D + e * 256 + tid] = v[e] * inv;
}

// ============ 2) S = relu(En @ En^T), diag = 1 (fp32 WMMA, 32x32/wave) ============
// 8 waves per block, each wave one 32x32 tile (2x2 WMMA fragments). 16384 tiles.
__global__ void k_gram(const float* __restrict__ En, float* __restrict__ S) {
    int wave = threadIdx.x >> 5;
    int lane = threadIdx.x & 31;
    int tile = blockIdx.x * 8 + wave;          // 0..16383
    int rt = tile >> 7, ct = tile & 127;       // 128 tiles per dim (32 wide)
    int m = lane & 15, half = lane >> 4;
    const float* pa0 = En + (size_t)(rt * 32 + m) * D + 2 * half;
    const float* pa1 = En + (size_t)(rt * 32 + 16 + m) * D + 2 * half;
    const float* pb0 = En + (size_t)(ct * 32 + m) * D + 2 * half;
    const float* pb1 = En + (size_t)(ct * 32 + 16 + m) * D + 2 * half;
    v8f c00 = {0.f,0.f,0.f,0.f,0.f,0.f,0.f,0.f};
    v8f c01 = c00, c10 = c00, c11 = c00;
    for (int kb = 0; kb < D; kb += 32) {       // 128B chunk per row
        if (kb + 32 < D) {
            __builtin_prefetch(pa0 + kb + 32, 0, 1);
            __builtin_prefetch(pa1 + kb + 32, 0, 1);
            __builtin_prefetch(pb0 + kb + 32, 0, 1);
            __builtin_prefetch(pb1 + kb + 32, 0, 1);
        }
        #pragma unroll
        for (int k = kb; k < kb + 32; k += 4) {
            v2f a0 = *(const v2f*)(pa0 + k);
            v2f a1 = *(const v2f*)(pa1 + k);
            v2f b0 = *(const v2f*)(pb0 + k);
            v2f b1 = *(const v2f*)(pb1 + k);
            c00 = __builtin_amdgcn_wmma_f32_16x16x4_f32(false, a0, false, b0, (short)0, c00, false, false);
            c01 = __builtin_amdgcn_wmma_f32_16x16x4_f32(false, a0, false, b1, (short)0, c01, false, false);
            c10 = __builtin_amdgcn_wmma_f32_16x16x4_f32(false, a1, false, b0, (short)0, c10, false, false);
            c11 = __builtin_amdgcn_wmma_f32_16x16x4_f32(false, a1, false, b1, (short)0, c11, false, false);
        }
    }
    int col0 = ct * 32 + m, col1 = col0 + 16;
    #pragma unroll
    for (int r = 0; r < 8; ++r) {
        int row0 = rt * 32 + r + half * 8;
        int row1 = row0 + 16;
        float v;
        v = fmaxf(c00[r], 0.f); if (row0 == col0) v = 1.0f; S[(size_t)row0 * N + col0] = v;
        v = fmaxf(c01[r], 0.f); if (row0 == col1) v = 1.0f; S[(size_t)row0 * N + col1] = v;
        v = fmaxf(c10[r], 0.f); if (row1 == col0) v = 1.0f; S[(size_t)row1 * N + col0] = v;
        v = fmaxf(c11[r], 0.f); if (row1 == col1) v = 1.0f; S[(size_t)row1 * N + col1] = v;
    }
}

// ============ 3) per-row 7th largest -> sigma ============
__device__ __forceinline__ void top7_insert(float* t, float v) {
    if (v > t[6]) {
        t[6] = v;
        #pragma unroll
        for (int q = 6; q > 0; --q) {
            if (t[q] > t[q - 1]) { float tmp = t[q - 1]; t[q - 1] = t[q]; t[q] = tmp; }
            else break;
        }
    }
}

__global__ void k_top7(const float* __restrict__ S, float* __restrict__ sig) {
    __shared__ float lds[256 * 7];
    int row = blockIdx.x, tid = threadIdx.x;
    float t[7];
    #pragma unroll
    for (int q = 0; q < 7; ++q) t[q] = -1e30f;
    for (int j = tid; j < N; j += 256)
        top7_insert(t, S[(size_t)row * N + j]);
    #pragma unroll
    for (int q = 0; q < 7; ++q) lds[tid * 7 + q] = t[q];
    __syncthreads();
    for (int s = 128; s >= 1; s >>= 1) {
        if (tid < s) {
            #pragma unroll
            for (int q = 0; q < 7; ++q) top7_insert(t, lds[(tid + s) * 7 + q]);
            #pragma unroll
            for (int q = 0; q < 7; ++q) lds[tid * 7 + q] = t[q];
        }
        __syncthreads();
    }
    if (tid == 0) sig[row] = fmaxf(t[6], 0.0f);   // relu, matches reference
}

// ============ 4) A = exp(-S^2/(si*sj)) (in place) + block partial sums ============
__global__ void k_rbf(float* __restrict__ A, const float* __restrict__ sig,
                      float* __restrict__ part) {
    __shared__ float red[256];
    int tid = threadIdx.x;
    size_t base = (size_t)blockIdx.x * 1024;
    float local = 0.f;
    #pragma unroll
    for (int e = 0; e < 4; ++e) {
        size_t idx = base + e * 256 + tid;
        int row = (int)(idx >> 12), col = (int)(idx & 4095);
        float s = A[idx];
        float a = expf(-(s * s) / (sig[row] * sig[col]));
        A[idx] = a;
        local += a;
    }
    red[tid] = local; __syncthreads();
    for (int s = 128; s >= 1; s >>= 1) {
        if (tid < s) red[tid] += red[tid + s];
        __syncthreads();
    }
    if (tid == 0) part[blockIdx.x] = red[0];
}

// ============ 5) reduce partials -> mean ============
__global__ void k_mean(const float* __restrict__ part, float* __restrict__ mean) {
    __shared__ float red[256];
    int tid = threadIdx.x;
    float s = 0.f;
    for (int i = tid; i < 16384; i += 256) s += part[i];
    red[tid] = s; __syncthreads();
    for (int t = 128; t >= 1; t >>= 1) {
        if (tid < t) red[tid] += red[tid + t];
        __syncthreads();
    }
    if (tid == 0) *mean = red[0] * (1.0f / ((float)N * (float)N));
}

// ============ 6) threshold A at mean ============
__global__ void k_thresh(float* __restrict__ A, const float* __restrict__ meanp) {
    float mean = *meanp;
    int tid = threadIdx.x;
    size_t base = (size_t)blockIdx.x * 1024;
    #pragma unroll
    for (int e = 0; e < 4; ++e) {
        size_t idx = base + e * 256 + tid;
        float a = A[idx];
        A[idx] = (a < mean) ? 0.0f : a;
    }
}

// ============ 7) init X0 ============
__global__ void k_initx(const int* __restrict__ labels, float* __restrict__ X) {
    int idx = blockIdx.x * 256 + threadIdx.x;
    if (idx >= N * C) return;
    int row = idx / C, col = idx - row * C;
    float v;
    if (row < NLAB) v = (labels[row] == col) ? 1.0f : 0.0f;
    else            v = 1.0f / (float)C;
    X[idx] = v;
}

// ============ 7b) XT[c][r] = X[r][c], zero-padded to 128 cols ============
__global__ void k_transpose(const float* __restrict__ X, float* __restrict__ XT) {
    int idx = blockIdx.x * 256 + threadIdx.x;    // over CPAD*N
    int c = idx >> 12;                           // idx / 4096
    int r = idx & 4095;
    XT[idx] = (c < C) ? X[(size_t)r * C + c] : 0.0f;
}

// ============ 8) Y = A @ X via X^T (fp32 WMMA, 16x32/wave) ============
// One wave per (row tile, 32-col group): 256 x 4 = 1024 waves.
__global__ void k_gtg_mm(const float* __restrict__ A, const float* __restrict__ XT,
                         float* __restrict__ Y) {
    int lane = threadIdx.x;
    int rt = blockIdx.x >> 2, ctg = blockIdx.x & 3;
    int m = lane & 15, half = lane >> 4;
    int cbase = ctg * 32;
    const float* pa  = A  + (size_t)(rt * 16 + m) * N + 2 * half;
    const float* pb0 = XT + (size_t)(cbase + m) * N + 2 * half;
    const float* pb1 = XT + (size_t)(cbase + 16 + m) * N + 2 * half;
    v8f c0 = {0.f,0.f,0.f,0.f,0.f,0.f,0.f,0.f};
    v8f c1 = c0;
    for (int kb = 0; kb < N; kb += 32) {
        if (kb + 32 < N) {
            __builtin_prefetch(pa  + kb + 32, 0, 1);
            __builtin_prefetch(pb0 + kb + 32, 0, 1);
            __builtin_prefetch(pb1 + kb + 32, 0, 1);
        }
        #pragma unroll
        for (int k = kb; k < kb + 32; k += 4) {
            v2f a  = *(const v2f*)(pa  + k);
            v2f b0 = *(const v2f*)(pb0 + k);
            v2f b1 = *(const v2f*)(pb1 + k);
            c0 = __builtin_amdgcn_wmma_f32_16x16x4_f32(false, a, false, b0, (short)0, c0, false, false);
            c1 = __builtin_amdgcn_wmma_f32_16x16x4_f32(false, a, false, b1, (short)0, c1, false, false);
        }
    }
    int col0 = cbase + m, col1 = col0 + 16;
    #pragma unroll
    for (int r = 0; r < 8; ++r) {
        int row = rt * 16 + r + half * 8;
        if (col0 < C) Y[(size_t)row * C + col0] = c0[r];
        if (col1 < C) Y[(size_t)row * C + col1] = c1[r];
    }
}

// ============ 9) X <- (X + X*Y) / (rowsum + eps) ============
__global__ void k_gtg_update(const float* __restrict__ X, const float* __restrict__ Y,
                             float* __restrict__ dst) {
    __shared__ float red[128];
    int row = blockIdx.x, tid = threadIdx.x;
    float mult = 0.f;
    if (tid < C) {
        float x = X[(size_t)row * C + tid];
        mult = x + x * Y[(size_t)row * C + tid];
    }
    red[tid] = mult; __syncthreads();
    for (int s = 64; s >= 1; s >>= 1) {
        if (tid < s) red[tid] += red[tid + s];
        __syncthreads();
    }
    float sum = red[0];
    if (tid < C) dst[(size_t)row * C + tid] = mult / (sum + EPS);
}

extern "C" void kernel_launch(void* const* d_in, const int* in_sizes, int n_in,
                              void* d_out, int out_size, void* d_ws, size_t ws_size,
                              hipStream_t stream) {
    const float* E      = (const float*)d_in[0];   // [4096, 1024]
    const int*   labels = (const int*)d_in[1];     // [4096]
    float* out = (float*)d_out;                    // [4096, 100]

    char* ws = (char*)d_ws;
    float* En   = (float*)(ws + OFF_EN);
    float* A    = (float*)(ws + OFF_A);
    float* sig  = (float*)(ws + OFF_SIG);
    float* part = (float*)(ws + OFF_PART);
    float* mean = (float*)(ws + OFF_MEAN);
    float* X    = (float*)(ws + OFF_X);
    float* Y    = (float*)(ws + OFF_Y);
    float* XT   = (float*)(ws + OFF_XT);

    k_normalize<<<N, 256, 0, stream>>>(E, En);
    k_gram<<<(N / 32) * (N / 32) / 8, 256, 0, stream>>>(En, A);   // 2048 blocks
    k_top7<<<N, 256, 0, stream>>>(A, sig);
    k_rbf<<<16384, 256, 0, stream>>>(A, sig, part);
    k_mean<<<1, 256, 0, stream>>>(part, mean);
    k_thresh<<<16384, 256, 0, stream>>>(A, mean);
    k_initx<<<(N * C + 255) / 256, 256, 0, stream>>>(labels, X);

    for (int it = 0; it < GTG_ITERS; ++it) {
        k_transpose<<<(CPAD * N) / 256, 256, 0, stream>>>(X, XT);
        k_gtg_mm<<<(N / 16) * 4, 32, 0, stream>>>(A, XT, Y);
        float* dst = (it == GTG_ITERS - 1) ? out : X;
        k_gtg_update<<<N, 128, 0, stream>>>(X, Y, dst);
    }
}